// KGNet_4252017623708
// MI455X (gfx1250) — compile-verified
//
#include <hip/hip_runtime.h>
#include <math.h>

// ---------------------------------------------------------------------------
// KGNet-style complex GRU block for MI455X (gfx1250), fp32 WMMA 16x16x4 path.
// Each wave of 32 lanes processes 16 batch samples; all small GEMMs run on the
// matrix unit via V_WMMA_F32_16X16X4_F32, elementwise GRU math on VALU.
// ---------------------------------------------------------------------------

typedef __attribute__((ext_vector_type(2))) float v2f;
typedef __attribute__((ext_vector_type(8))) float v8f;

constexpr int  NWAVES = 4;            // waves per block
constexpr int  WSZ    = 32;           // wave32
constexpr int  TPB    = NWAVES * WSZ; // 128 threads
constexpr long BATCH  = 524288;
constexpr int  KIN    = 9;            // input feature count
constexpr int  KINP   = 12;           // padded to multiple of 4
constexpr int  NFC    = 18;           // FC width
constexpr int  NH     = 18;           // hidden size
constexpr int  KP     = 20;           // K-padded activation stride (18 -> 20)
constexpr int  NG     = 54;           // 3*H gate width
constexpr int  GST    = 56;           // gate LDS row stride
constexpr int  NOUT   = 4;            // output L

struct WaveScratch {
    float x [16 * KINP]; // input staging (zero K-padded)
    float yr[16 * KP];   // real activation / later g_real
    float yi[16 * KP];   // imag activation / later g_imag
    float h [16 * KP];   // current GRU hidden state staging
    float gi[16 * GST];  // x-gates (raw); later reused as mr [16][KP]
    float gh[16 * GST];  // h-gates (raw); later reused as mi [16][KP]
};

// ---- WMMA fragment builders ------------------------------------------------
// A fragment (16x4 f32): lane m = lane&15 holds row m; VGPR pair holds
// K = k0+2*(lane>>4) and K+1.  abuf is LDS [16][astride], astride even.
__device__ __forceinline__ v2f afrag(const float* abuf, int astride, int k0, int lane) {
    int m  = lane & 15;
    int k  = k0 + 2 * (lane >> 4);
    return *(const v2f*)(abuf + m * astride + k);
}

// B fragment (4x16 f32) for Y = X @ W^T: B[k][n] = W[n][k].
// Read directly from global W[N][K] with zero padding for n>=N / k>=K.
__device__ __forceinline__ v2f bfrag(const float* W, int N, int K, int nb, int k0, int lane) {
    int n = nb + (lane & 15);
    int k = k0 + 2 * (lane >> 4);
    v2f r; r.x = 0.0f; r.y = 0.0f;
    if (n < N) {
        if (k     < K) r.x = W[n * K + k];
        if (k + 1 < K) r.y = W[n * K + k + 1];
    }
    return r;
}

// Accumulate one 16x16 output tile over ksteps K-chunks of 4.
__device__ __forceinline__ v8f wmma_kloop(const float* abuf, int astride,
                                          const float* W, int N, int K, int nb,
                                          int ksteps, int lane) {
    v8f acc = {0.f, 0.f, 0.f, 0.f, 0.f, 0.f, 0.f, 0.f};
#pragma unroll
    for (int s = 0; s < ksteps; ++s) {
        v2f a = afrag(abuf, astride, 4 * s, lane);
        v2f b = bfrag(W, N, K, nb, 4 * s, lane);
        acc = __builtin_amdgcn_wmma_f32_16x16x4_f32(
            /*neg_a=*/false, a, /*neg_b=*/false, b,
            /*c_mod=*/(short)0, acc, /*reuse_a=*/false, /*reuse_b=*/false);
    }
    return acc;
}

// Y[16,Nout] = prelu(X @ W^T + bias); result stored to ybuf [16][KP] with
// zeroed K-padding columns (ready to be the next stage's A operand).
__device__ __forceinline__ void fc_prelu(const float* xbuf, int astride, int ksteps,
                                         const float* W, int Nout, int K,
                                         const float* bias, float alpha,
                                         float* ybuf, int lane) {
    int col = lane & 15, half = lane >> 4;
    for (int t = 0; t * 16 < Nout; ++t) {
        v8f acc = wmma_kloop(xbuf, astride, W, Nout, K, 16 * t, ksteps, lane);
        int gc = 16 * t + col;
        if (gc < Nout) {
            float b = bias[gc];
#pragma unroll
            for (int v = 0; v < 8; ++v) {
                float y = acc[v] + b;
                y = (y >= 0.0f) ? y : alpha * y;
                ybuf[(v + 8 * half) * KP + gc] = y;
            }
        }
    }
    if (lane < 16) {                    // zero K-pad columns 18,19
        ybuf[lane * KP + NH]     = 0.0f;
        ybuf[lane * KP + NH + 1] = 0.0f;
    }
}

// Raw gate GEMM: gbuf[16][GST] = X[16][18] @ W[54][18]^T   (bias added later)
__device__ __forceinline__ void gate_mm(const float* xbuf, const float* W,
                                        float* gbuf, int lane) {
    int col = lane & 15, half = lane >> 4;
#pragma unroll
    for (int t = 0; t < 4; ++t) {
        v8f acc = wmma_kloop(xbuf, KP, W, NG, NH, 16 * t, 5, lane);
        int gc = 16 * t + col;
        if (gc < NG) {
#pragma unroll
            for (int v = 0; v < 8; ++v)
                gbuf[(v + 8 * half) * GST + gc] = acc[v];
        }
    }
}

__device__ __forceinline__ float sigmoidf_(float x) {
    return 1.0f / (1.0f + expf(-x));
}

// One GRU cell for 16 samples; writes new hidden state F straight to global.
__device__ __forceinline__ void gru_cell(const float* xbuf, float* hbuf,
                                         const float* Wih, const float* Whh,
                                         const float* bih, const float* bhh,
                                         const float* hglob, float* fglob,
                                         float* gibuf, float* ghbuf,
                                         long base, int lane) {
    // stage hidden state, zero K-padded
    for (int i = lane; i < 16 * KP; i += WSZ) {
        int r = i / KP, c = i % KP;
        hbuf[i] = (c < NH) ? hglob[(base + r) * NH + c] : 0.0f;
    }
    gate_mm(xbuf, Wih, gibuf, lane);
    gate_mm(hbuf, Whh, ghbuf, lane);
    // scalar gate phase: 16 samples x 18 units = 288 = 9 iters/lane
    for (int p = lane; p < 16 * NH; p += WSZ) {
        int m = p / NH, j = p % NH;
        float ir  = gibuf[m * GST + j]          + bih[j];
        float iz  = gibuf[m * GST + NH + j]     + bih[NH + j];
        float inn = gibuf[m * GST + 2 * NH + j] + bih[2 * NH + j];
        float hr  = ghbuf[m * GST + j]          + bhh[j];
        float hz  = ghbuf[m * GST + NH + j]     + bhh[NH + j];
        float hn  = ghbuf[m * GST + 2 * NH + j] + bhh[2 * NH + j];
        float r = sigmoidf_(ir + hr);
        float z = sigmoidf_(iz + hz);
        float n = tanhf(inn + r * hn);
        float f = (1.0f - z) * n + z * hbuf[m * KP + j];
        fglob[(base + m) * NH + j] = f;
    }
}

// Final 18->4 FC, direct global store.
__device__ __forceinline__ void out_fc(const float* mbuf, const float* Wo,
                                       const float* bo, float* kout,
                                       long base, int lane) {
    v8f acc = wmma_kloop(mbuf, KP, Wo, NOUT, NFC, 0, 5, lane);
    int col = lane & 15, half = lane >> 4;
    if (col < NOUT) {
        float b = bo[col];
#pragma unroll
        for (int v = 0; v < 8; ++v)
            kout[(base + v + 8 * half) * NOUT + col] = acc[v] + b;
    }
}

__global__ __launch_bounds__(TPB) void kgnet_gru_wmma_kernel(
    const float* __restrict__ xr,   const float* __restrict__ xi,
    const float* __restrict__ hrr,  const float* __restrict__ hir,
    const float* __restrict__ hri,  const float* __restrict__ hii,
    const float* __restrict__ WrIn, const float* __restrict__ brIn,
    const float* __restrict__ WiIn, const float* __restrict__ biIn,
    const float* __restrict__ a1p,
    const float* __restrict__ grWih, const float* __restrict__ grWhh,
    const float* __restrict__ grBih, const float* __restrict__ grBhh,
    const float* __restrict__ giWih, const float* __restrict__ giWhh,
    const float* __restrict__ giBih, const float* __restrict__ giBhh,
    const float* __restrict__ WrM,  const float* __restrict__ brM,
    const float* __restrict__ WiM,  const float* __restrict__ biM,
    const float* __restrict__ a2p,
    const float* __restrict__ WrO,  const float* __restrict__ brO,
    const float* __restrict__ WiO,  const float* __restrict__ biO,
    float* __restrict__ out)
{
    __shared__ WaveScratch sw[NWAVES];
    const int tid  = threadIdx.x;
    const int wave = tid >> 5;
    const int lane = tid & 31;
    WaveScratch& S = sw[wave];

    const long tile = (long)blockIdx.x * NWAVES + wave;
    const long base = tile * 16;            // first sample of this wave's tile

    const float a1 = a1p[0];
    const float a2 = a2p[0];

    // output layout: kr(B*4) | ki(B*4) | Frr(B*18) | Fir | Fri | Fii
    float* o_kr  = out;
    float* o_ki  = out + 4L  * BATCH;
    float* o_frr = out + 8L  * BATCH;
    float* o_fir = out + 26L * BATCH;
    float* o_fri = out + 44L * BATCH;
    float* o_fii = out + 62L * BATCH;

    // ---- input FC + PReLU (real) ----
    for (int i = lane; i < 16 * KINP; i += WSZ) {
        int r = i / KINP, c = i % KINP;
        S.x[i] = (c < KIN) ? xr[(base + r) * KIN + c] : 0.0f;
    }
    fc_prelu(S.x, KINP, 3, WrIn, NFC, KIN, brIn, a1, S.yr, lane);

    // ---- input FC + PReLU (imag) ----
    for (int i = lane; i < 16 * KINP; i += WSZ) {
        int r = i / KINP, c = i % KINP;
        S.x[i] = (c < KIN) ? xi[(base + r) * KIN + c] : 0.0f;
    }
    fc_prelu(S.x, KINP, 3, WiIn, NFC, KIN, biIn, a1, S.yi, lane);

    // ---- four GRU cells (Frr, Fir, Fri, Fii) ----
    gru_cell(S.yr, S.h, grWih, grWhh, grBih, grBhh, hrr, o_frr, S.gi, S.gh, base, lane);
    gru_cell(S.yi, S.h, grWih, grWhh, grBih, grBhh, hir, o_fir, S.gi, S.gh, base, lane);
    gru_cell(S.yr, S.h, giWih, giWhh, giBih, giBhh, hri, o_fri, S.gi, S.gh, base, lane);
    gru_cell(S.yi, S.h, giWih, giWhh, giBih, giBhh, hii, o_fii, S.gi, S.gh, base, lane);

    // ---- complex combine: g_real -> yr, g_imag -> yi (K-padded) ----
    // (same-wave global store->load to o_f* is order-preserved)
    for (int p = lane; p < 16 * KP; p += WSZ) {
        int m = p / KP, c = p % KP;
        float gre = 0.0f, gim = 0.0f;
        if (c < NH) {
            long off = (base + m) * NH + c;
            gre = o_frr[off] - o_fii[off];
            gim = o_fri[off] + o_fir[off];
        }
        S.yr[p] = gre;
        S.yi[p] = gim;
    }

    // ---- mid FC + PReLU (reuse gate buffers as [16][KP] staging) ----
    float* mr = S.gi;
    float* mi = S.gh;
    fc_prelu(S.yr, KP, 5, WrM, NFC, NH, brM, a2, mr, lane);
    fc_prelu(S.yi, KP, 5, WiM, NFC, NH, biM, a2, mi, lane);

    // ---- output FC ----
    out_fc(mr, WrO, brO, o_kr, base, lane);
    out_fc(mi, WiO, biO, o_ki, base, lane);
}

extern "C" void kernel_launch(void* const* d_in, const int* in_sizes, int n_in,
                              void* d_out, int out_size, void* d_ws, size_t ws_size,
                              hipStream_t stream) {
    (void)in_sizes; (void)n_in; (void)out_size; (void)d_ws; (void)ws_size;
    const float* xr    = (const float*)d_in[0];
    const float* xi    = (const float*)d_in[1];
    const float* hrr   = (const float*)d_in[2];
    const float* hir   = (const float*)d_in[3];
    const float* hri   = (const float*)d_in[4];
    const float* hii   = (const float*)d_in[5];
    const float* WrIn  = (const float*)d_in[6];
    const float* brIn  = (const float*)d_in[7];
    const float* WiIn  = (const float*)d_in[8];
    const float* biIn  = (const float*)d_in[9];
    const float* a1p   = (const float*)d_in[10];
    const float* grWih = (const float*)d_in[11];
    const float* grWhh = (const float*)d_in[12];
    const float* grBih = (const float*)d_in[13];
    const float* grBhh = (const float*)d_in[14];
    const float* giWih = (const float*)d_in[15];
    const float* giWhh = (const float*)d_in[16];
    const float* giBih = (const float*)d_in[17];
    const float* giBhh = (const float*)d_in[18];
    const float* WrM   = (const float*)d_in[19];
    const float* brM   = (const float*)d_in[20];
    const float* WiM   = (const float*)d_in[21];
    const float* biM   = (const float*)d_in[22];
    const float* a2p   = (const float*)d_in[23];
    const float* WrO   = (const float*)d_in[24];
    const float* brO   = (const float*)d_in[25];
    const float* WiO   = (const float*)d_in[26];
    const float* biO   = (const float*)d_in[27];

    dim3 grid((unsigned)(BATCH / (16 * NWAVES)));   // 8192 blocks
    dim3 block(TPB);                                // 128 threads = 4 waves
    kgnet_gru_wmma_kernel<<<grid, block, 0, stream>>>(
        xr, xi, hrr, hir, hri, hii,
        WrIn, brIn, WiIn, biIn, a1p,
        grWih, grWhh, grBih, grBhh,
        giWih, giWhh, giBih, giBhh,
        WrM, brM, WiM, biM, a2p,
        WrO, brO, WiO, biO,
        (float*)d_out);
}